// SofaNetEllipse_66331474919980
// MI455X (gfx1250) — compile-verified
//
#include <hip/hip_runtime.h>
#include <math.h>

typedef __attribute__((ext_vector_type(2))) float v2f;
typedef __attribute__((ext_vector_type(8))) float v8f;

#define HID 256
#define TILE_M 16
#define SA_STRIDE 260   // 256+4; stride % 64 == 4 -> conflict-free WMMA operand reads
#define PI_F 3.14159265358979323846f

struct Params {
  const float* t;
  const float* W0[3]; const float* b0[3];
  const float* W1[3]; const float* b1[3];
  const float* W2[3]; const float* b2[3];
  float* out;
  int N;
};

__device__ __forceinline__ float fast_tanh(float x) {
  // tanh(x) = 1 - 2/(exp(2x)+1); saturates correctly at +-inf
  float e = __expf(2.0f * x);
  return 1.0f - 2.0f * __builtin_amdgcn_rcpf(e + 1.0f);
}

__global__ __launch_bounds__(128)
void sofa_net_ellipse_kernel(Params p) {
  __shared__ float sA [TILE_M][SA_STRIDE];   // activations (primal)
  __shared__ float sAd[TILE_M][SA_STRIDE];   // activations (tangent)
  __shared__ float tsh[TILE_M];
  __shared__ float pP[4][32];
  __shared__ float pD[4][32];
  __shared__ float resP[3][2][TILE_M];
  __shared__ float resD[3][2][TILE_M];

  const int tid  = threadIdx.x;
  const int lane = tid & 31;
  const int wave = tid >> 5;
  const int kh   = lane >> 4;   // half-wave selector (K split for A/B operands)
  const int ln   = lane & 15;   // M for A, N for B/D

  const int row0 = blockIdx.x * TILE_M;

  if (tid < TILE_M) {
    int row = row0 + tid;
    tsh[tid] = (row < p.N) ? p.t[row] : 0.0f;
  }
  __syncthreads();

  for (int net = 0; net < 3; ++net) {
    const float* __restrict__ W0 = p.W0[net]; const float* __restrict__ b0 = p.b0[net];
    const float* __restrict__ W1 = p.W1[net]; const float* __restrict__ b1 = p.b1[net];
    const float* __restrict__ W2 = p.W2[net]; const float* __restrict__ b2 = p.b2[net];

    // ---- layer 0: h0 = tanh(W0*u + b0), h0d = (1-h0^2)*W0*pi   (u = pi*t, du/dt = pi)
    for (int idx = tid; idx < TILE_M * HID; idx += 128) {
      int r = idx >> 8;
      int c = idx & (HID - 1);
      float u = PI_F * tsh[r];
      float w = W0[c];
      float h = fast_tanh(w * u + b0[c]);
      sA [r][c] = h;
      sAd[r][c] = (1.0f - h * h) * (w * PI_F);
    }
    __syncthreads();

    // ---- layer 1: z1 = h0 @ W1^T + b1 (and tangent), via V_WMMA_F32_16X16X4_F32
    // wave owns 4 chunks of 16 output features; K-loop of 64 wmma steps each.
    v8f hP[4], hD[4];
    #pragma unroll
    for (int i = 0; i < 4; ++i) {
      const int n0 = (wave * 4 + i) * 16;
      const float bv = b1[n0 + ln];         // bias depends on N = ln only
      v8f acc, accd;
      #pragma unroll
      for (int g = 0; g < 8; ++g) { acc[g] = bv; accd[g] = 0.0f; }

      const float* __restrict__ Wrow = W1 + (n0 + ln) * HID;  // B[k][n] = W1[n][k]
      #pragma unroll 4
      for (int k0 = 0; k0 < HID; k0 += 4) {
        const int koff = k0 + 2 * kh;       // A/B: VGPR g holds K = koff + g
        v2f a    = *reinterpret_cast<const v2f*>(&sA [ln][koff]);
        v2f ad   = *reinterpret_cast<const v2f*>(&sAd[ln][koff]);
        v2f bmat = *reinterpret_cast<const v2f*>(Wrow + koff); // shared by both WMMAs
        acc  = __builtin_amdgcn_wmma_f32_16x16x4_f32(false, a,  false, bmat,
                                                     (short)0, acc,  false, false);
        accd = __builtin_amdgcn_wmma_f32_16x16x4_f32(false, ad, false, bmat,
                                                     (short)0, accd, false, false);
      }
      #pragma unroll
      for (int g = 0; g < 8; ++g) {
        float h = fast_tanh(acc[g]);
        hP[i][g] = h;
        hD[i][g] = (1.0f - h * h) * accd[g];
      }
    }
    __syncthreads();   // everyone done reading h0 before overwrite

    // write h1/h1d back: D layout is M = g + 8*kh, N = ln
    #pragma unroll
    for (int i = 0; i < 4; ++i) {
      const int n0 = (wave * 4 + i) * 16;
      #pragma unroll
      for (int g = 0; g < 8; ++g) {
        int m = g + 8 * kh;
        sA [m][n0 + ln] = hP[i][g];
        sAd[m][n0 + ln] = hD[i][g];
      }
    }
    __syncthreads();

    // ---- layer 2: 2 outputs/row, VALU reduction (2x256 per row is negligible)
    {
      int r = tid & 15;
      int o = (tid >> 4) & 1;
      int q = wave;                          // K quarter
      const float* __restrict__ w2 = W2 + o * HID;
      float s = 0.0f, sd = 0.0f;
      #pragma unroll 4
      for (int k = q * 64; k < q * 64 + 64; ++k) {
        float w = w2[k];
        s  += sA [r][k] * w;
        sd += sAd[r][k] * w;
      }
      pP[q][lane] = s;
      pD[q][lane] = sd;
    }
    __syncthreads();
    if (tid < 32) {
      int r = tid & 15;
      int o = tid >> 4;
      resP[net][o][r] = pP[0][tid] + pP[1][tid] + pP[2][tid] + pP[3][tid] + b2[o];
      resD[net][o][r] = pD[0][tid] + pD[1][tid] + pD[2][tid] + pD[3][tid];
    }
    __syncthreads();
  }

  // ---- combine: product rule for xp, yp and their tangents
  if (tid < TILE_M && row0 + tid < p.N) {
    int r = tid;
    float t = tsh[r];
    float u = PI_F * t;
    float a  = resP[0][0][r], b_ = resP[0][1][r];
    float ad = resD[0][0][r], bd = resD[0][1][r];
    float kx = resP[1][0][r], ky = resP[1][1][r];
    float kxd= resD[1][0][r], kyd= resD[1][1][r];
    float sx = resP[2][0][r], sy = resP[2][1][r];
    float sxd= resD[2][0][r], syd= resD[2][1][r];

    float cu = cosf(u), su = sinf(u);
    float xp0  = a * (cu - 1.0f);
    float xp0d = ad * (cu - 1.0f) - a * su * PI_F;
    float yp0  = b_ * su;
    float yp0d = bd * su + b_ * cu * PI_F;

    float xp  = xp0 * kx + sx;
    float xpd = xp0d * kx + xp0 * kxd + sxd;
    float yp  = yp0 * ky + sy;
    float ypd = yp0d * ky + yp0 * kyd + syd;

    int row = row0 + r;
    int N = p.N;
    p.out[0 * N + row] = t * (PI_F * 0.5f);   // alpha
    p.out[1 * N + row] = xp;
    p.out[2 * N + row] = yp;
    p.out[3 * N + row] = PI_F * 0.5f;         // alpha_prime
    p.out[4 * N + row] = xpd;
    p.out[5 * N + row] = ypd;
  }
}

extern "C" void kernel_launch(void* const* d_in, const int* in_sizes, int n_in,
                              void* d_out, int out_size, void* d_ws, size_t ws_size,
                              hipStream_t stream) {
  (void)n_in; (void)out_size; (void)d_ws; (void)ws_size;
  Params p;
  p.t = (const float*)d_in[0];
  for (int net = 0; net < 3; ++net) {
    int base = 1 + net * 6;   // order: W0,b0,W1,b1,W2,b2 per net (ab, kxy, sxy)
    p.W0[net] = (const float*)d_in[base + 0];
    p.b0[net] = (const float*)d_in[base + 1];
    p.W1[net] = (const float*)d_in[base + 2];
    p.b1[net] = (const float*)d_in[base + 3];
    p.W2[net] = (const float*)d_in[base + 4];
    p.b2[net] = (const float*)d_in[base + 5];
  }
  p.out = (float*)d_out;
  p.N = in_sizes[0];
  int blocks = (p.N + TILE_M - 1) / TILE_M;
  sofa_net_ellipse_kernel<<<blocks, 128, 0, stream>>>(p);
}